// GCN_14783277433401
// MI455X (gfx1250) — compile-verified
//
#include <hip/hip_runtime.h>
#include <math.h>

#define N_NODES  100000
#define N_EDGES  1600000
#define N_GRAPHS 100
#define EDGE_DIM 4
#define D_HID    128
#define KDIM     132   // D_HID + EDGE_DIM
#define D_OUT    4

typedef __attribute__((ext_vector_type(2))) float v2f;
typedef __attribute__((ext_vector_type(8))) float v8f;

// ---------------------------------------------------------------------------
// agg[i, 0:128] = x[i]  (self loop), agg[i, 128:132] = 0 (self-loop edge attr)
// ---------------------------------------------------------------------------
__global__ __launch_bounds__(256) void gnn_init_agg(const float* __restrict__ x,
                                                    float* __restrict__ agg) {
  size_t idx = (size_t)blockIdx.x * 256 + threadIdx.x;
  if (idx >= (size_t)N_NODES * KDIM) return;
  int i = (int)(idx / KDIM);
  int c = (int)(idx % KDIM);
  agg[idx] = (c < D_HID) ? x[(size_t)i * D_HID + c] : 0.0f;
}

// ---------------------------------------------------------------------------
// One wave per edge: lanes cover the 128-float row as float4; L2-resident
// global_atomic_add_f32 scatter (random dst). Lanes 0..3 add the edge attrs.
// ---------------------------------------------------------------------------
__global__ __launch_bounds__(256) void gnn_edge_scatter(
    const int* __restrict__ src, const int* __restrict__ dst,
    const float* __restrict__ ea, const float* __restrict__ x,
    float* __restrict__ agg) {
  int e = blockIdx.x * 8 + (threadIdx.x >> 5);   // grid exactly covers N_EDGES
  int lane = threadIdx.x & 31;
  int s = src[e];
  int d = dst[e];
  const float4 v = ((const float4*)(x + (size_t)s * D_HID))[lane];
  float* ag = agg + (size_t)d * KDIM + lane * 4;
  atomicAdd(ag + 0, v.x);
  atomicAdd(ag + 1, v.y);
  atomicAdd(ag + 2, v.z);
  atomicAdd(ag + 3, v.w);
  if (lane < EDGE_DIM)
    atomicAdd(agg + (size_t)d * KDIM + D_HID + lane,
              ea[(size_t)e * EDGE_DIM + lane]);
}

// ---------------------------------------------------------------------------
// x' = relu(agg[N,132] @ W[132,128] + b) via V_WMMA_F32_16X16X4_F32.
// One wave -> 16 rows x 128 cols (8 accumulators), A frag reused per K-chunk.
// W is staged transposed in (dynamic) LDS so B frags are contiguous b64 loads.
// ---------------------------------------------------------------------------
__global__ __launch_bounds__(256) void gnn_gemm_relu(
    const float* __restrict__ agg, const float* __restrict__ Wmat,
    const float* __restrict__ bias, float* __restrict__ out) {
  extern __shared__ float smem[];
  float* sWt = smem;                 // [128][132]  (transposed W)
  float* sB  = smem + D_HID * KDIM;  // [128]
  for (int idx = threadIdx.x; idx < KDIM * D_HID; idx += 256) {
    int k = idx / D_HID, n = idx % D_HID;
    sWt[n * KDIM + k] = Wmat[idx];
  }
  if (threadIdx.x < D_HID) sB[threadIdx.x] = bias[threadIdx.x];
  __syncthreads();

  int wave = blockIdx.x * 8 + (threadIdx.x >> 5);
  int lane = threadIdx.x & 31;
  int m0 = wave * 16;
  if (m0 >= N_NODES) return;
  int lo = lane & 15;
  int hi = lane >> 4;

  v8f acc[8] = {};  // zero-init accumulators

  // A-frag addressing: lane holds agg[m0+lo][k0 + 2*hi .. +1]  (8B aligned)
  const float* arow = agg + (size_t)(m0 + lo) * KDIM + 2 * hi;
  for (int k0 = 0; k0 < KDIM; k0 += 4) {
    v2f a = *(const v2f*)(arow + k0);
#pragma unroll
    for (int n = 0; n < 8; ++n) {
      // B-frag: W[k0+2*hi][col], W[k0+2*hi+1][col] -> contiguous in sWt
      v2f b = *(const v2f*)(sWt + (size_t)(n * 16 + lo) * KDIM + k0 + 2 * hi);
      acc[n] = __builtin_amdgcn_wmma_f32_16x16x4_f32(
          /*neg_a=*/false, a, /*neg_b=*/false, b,
          /*c_mod=*/(short)0, acc[n], /*reuse_a=*/false, /*reuse_b=*/false);
    }
  }

  // C/D layout: vgpr r, lanes 0-15 -> M=r, lanes 16-31 -> M=r+8; N=lo
#pragma unroll
  for (int n = 0; n < 8; ++n) {
    int col = n * 16 + lo;
    float bv = sB[col];
#pragma unroll
    for (int r = 0; r < 8; ++r) {
      float v = acc[n][r] + bv;
      out[(size_t)(m0 + r + 8 * hi) * D_HID + col] = fmaxf(v, 0.0f);
    }
  }
}

// ---------------------------------------------------------------------------
// Pooling: zero then atomic scatter-mean pieces
// ---------------------------------------------------------------------------
__global__ __launch_bounds__(256) void gnn_zero_pool(float* pooled, float* counts) {
  int idx = blockIdx.x * 256 + threadIdx.x;
  if (idx < N_GRAPHS * D_HID) pooled[idx] = 0.0f;
  if (idx < N_GRAPHS) counts[idx] = 0.0f;
}

__global__ __launch_bounds__(256) void gnn_pool(const float* __restrict__ x,
                                                const int* __restrict__ batch,
                                                float* pooled, float* counts) {
  size_t idx = (size_t)blockIdx.x * 256 + threadIdx.x;  // over N_NODES*128
  int i = (int)(idx >> 7);
  int c = (int)(idx & 127);
  int g = batch[i];
  atomicAdd(&pooled[(size_t)g * D_HID + c], x[idx]);
  if (c == 0) atomicAdd(&counts[g], 1.0f);
}

// ---------------------------------------------------------------------------
// Head: pooled/cnt @ Wout + bout, then log_softmax. One thread per graph.
// ---------------------------------------------------------------------------
__global__ void gnn_head(const float* __restrict__ pooled,
                         const float* __restrict__ counts,
                         const float* __restrict__ Wout,
                         const float* __restrict__ bout,
                         float* __restrict__ out) {
  int g = threadIdx.x;
  if (g >= N_GRAPHS) return;
  float inv = 1.0f / fmaxf(counts[g], 1.0f);
  float l[D_OUT];
#pragma unroll
  for (int o = 0; o < D_OUT; ++o) l[o] = bout[o];
  for (int c = 0; c < D_HID; ++c) {
    float p = pooled[(size_t)g * D_HID + c] * inv;
#pragma unroll
    for (int o = 0; o < D_OUT; ++o) l[o] += p * Wout[c * D_OUT + o];
  }
  float m = l[0];
#pragma unroll
  for (int o = 1; o < D_OUT; ++o) m = fmaxf(m, l[o]);
  float s = 0.0f;
#pragma unroll
  for (int o = 0; o < D_OUT; ++o) s += __expf(l[o] - m);
  float lse = m + __logf(s);
#pragma unroll
  for (int o = 0; o < D_OUT; ++o) out[g * D_OUT + o] = l[o] - lse;
}

// ---------------------------------------------------------------------------
extern "C" void kernel_launch(void* const* d_in, const int* in_sizes, int n_in,
                              void* d_out, int out_size, void* d_ws, size_t ws_size,
                              hipStream_t stream) {
  const float* x_in       = (const float*)d_in[0];
  const int*   edge_index = (const int*)d_in[1];   // [2, E]
  const float* edge_attr  = (const float*)d_in[2];
  const int*   batch      = (const int*)d_in[3];
  const float* Wl[3] = {(const float*)d_in[4], (const float*)d_in[6], (const float*)d_in[8]};
  const float* Bl[3] = {(const float*)d_in[5], (const float*)d_in[7], (const float*)d_in[9]};
  const float* Wout = (const float*)d_in[10];
  const float* bout = (const float*)d_in[11];

  // workspace carve-up (~104 MB)
  float* agg    = (float*)d_ws;                          // N*132
  float* xbuf   = agg + (size_t)N_NODES * KDIM;          // N*128 (16B aligned)
  float* pooled = xbuf + (size_t)N_NODES * D_HID;        // 100*128
  float* counts = pooled + N_GRAPHS * D_HID;             // 100

  const int* esrc = edge_index;
  const int* edst = edge_index + N_EDGES;

  const size_t lds_bytes = (size_t)(KDIM * D_HID + D_HID) * sizeof(float); // ~68KB

  const float* xcur = x_in;
  for (int l = 0; l < 3; ++l) {
    gnn_init_agg<<<(N_NODES * KDIM + 255) / 256, 256, 0, stream>>>(xcur, agg);
    gnn_edge_scatter<<<N_EDGES / 8, 256, 0, stream>>>(esrc, edst, edge_attr, xcur, agg);
    gnn_gemm_relu<<<(N_NODES / 16 + 7) / 8, 256, lds_bytes, stream>>>(agg, Wl[l], Bl[l], xbuf);
    xcur = xbuf;
  }

  gnn_zero_pool<<<(N_GRAPHS * D_HID + N_GRAPHS + 255) / 256, 256, 0, stream>>>(pooled, counts);
  gnn_pool<<<(N_NODES * D_HID) / 256, 256, 0, stream>>>(xbuf, batch, pooled, counts);
  gnn_head<<<1, 128, 0, stream>>>(pooled, counts, Wout, bout, (float*)d_out);
}